// Frames_Base_65231963292425
// MI455X (gfx1250) — compile-verified
//
#include <hip/hip_runtime.h>
#include <hip/hip_bf16.h>
#include <math.h>

typedef float v2f __attribute__((ext_vector_type(2)));
typedef float v8f __attribute__((ext_vector_type(8)));

#define BATCH 8
#define NPTS  4096
#define KNN   32
#define WAVES_PER_BLOCK 4
#define BLOCK_THREADS   (WAVES_PER_BLOCK * 32)

// Jacobi rotation on symmetric 3x3 Am with accumulated eigenvectors Vm.
// p,q are literal constants -> everything stays in registers.
#define JROTA(p, q)                                                            \
  do {                                                                         \
    float apq = Am[p][q];                                                      \
    if (fabsf(apq) > 1e-30f) {                                                 \
      float theta = (Am[q][q] - Am[p][p]) / (2.0f * apq);                      \
      float t = 1.0f / (fabsf(theta) + sqrtf(theta * theta + 1.0f));           \
      t = (theta < 0.0f) ? -t : t;                                             \
      float cc = 1.0f / sqrtf(t * t + 1.0f);                                   \
      float ss = t * cc;                                                       \
      float t1, t2;                                                            \
      t1 = Am[0][p]; t2 = Am[0][q]; Am[0][p] = cc*t1 - ss*t2; Am[0][q] = ss*t1 + cc*t2; \
      t1 = Am[1][p]; t2 = Am[1][q]; Am[1][p] = cc*t1 - ss*t2; Am[1][q] = ss*t1 + cc*t2; \
      t1 = Am[2][p]; t2 = Am[2][q]; Am[2][p] = cc*t1 - ss*t2; Am[2][q] = ss*t1 + cc*t2; \
      t1 = Am[p][0]; t2 = Am[q][0]; Am[p][0] = cc*t1 - ss*t2; Am[q][0] = ss*t1 + cc*t2; \
      t1 = Am[p][1]; t2 = Am[q][1]; Am[p][1] = cc*t1 - ss*t2; Am[q][1] = ss*t1 + cc*t2; \
      t1 = Am[p][2]; t2 = Am[q][2]; Am[p][2] = cc*t1 - ss*t2; Am[q][2] = ss*t1 + cc*t2; \
      t1 = Vm[0][p]; t2 = Vm[0][q]; Vm[0][p] = cc*t1 - ss*t2; Vm[0][q] = ss*t1 + cc*t2; \
      t1 = Vm[1][p]; t2 = Vm[1][q]; Vm[1][p] = cc*t1 - ss*t2; Vm[1][q] = ss*t1 + cc*t2; \
      t1 = Vm[2][p]; t2 = Vm[2][q]; Vm[2][p] = cc*t1 - ss*t2; Vm[2][q] = ss*t1 + cc*t2; \
    }                                                                          \
  } while (0)

__device__ __forceinline__ float pick3(float a, float b, float c, int j) {
  return (j == 0) ? a : ((j == 1) ? b : c);
}

__global__ __launch_bounds__(BLOCK_THREADS) void
knn_frames_kernel(const float* __restrict__ pnts,
                  float* __restrict__ F_out,
                  float* __restrict__ C_out,
                  float* __restrict__ PC_out) {
  // Per-wave LDS: top-k lists (lane-private columns) + distance-tile transpose.
  __shared__ unsigned long long topk[WAVES_PER_BLOCK][KNN][32];   // 32 KB
  __shared__ float dtile[WAVES_PER_BLOCK][32][17];                // padded, ~8.7 KB

  const int tid  = threadIdx.x;
  const int w    = tid >> 5;
  const int lane = tid & 31;

  const int gpt = blockIdx.x * BLOCK_THREADS + tid;   // global point id
  const int b   = gpt / NPTS;
  const int q   = gpt - b * NPTS;                     // query index in batch
  const int q0  = q - lane;                           // wave's first query

  const float* __restrict__ bp = pnts + (size_t)b * NPTS * 3;

  // ---- load this lane's query point ----
  const float qx = bp[(size_t)q * 3 + 0];
  const float qy = bp[(size_t)q * 3 + 1];
  const float qz = bp[(size_t)q * 3 + 2];
  const float q2 = qx * qx + qy * qy + qz * qz;

  const int  m  = lane & 15;
  const bool hi = lane >= 16;

  // ---- build A matrices (16x4 f32 each): A0 = queries 0-15, A1 = 16-31 ----
  // Layout (ISA 7.12.2): V0 = {K0 | K2}, V1 = {K1 | K3} across lane halves.
  float sx0 = __shfl(qx, m, 32),      sy0 = __shfl(qy, m, 32),      sz0 = __shfl(qz, m, 32);
  float sx1 = __shfl(qx, m + 16, 32), sy1 = __shfl(qy, m + 16, 32), sz1 = __shfl(qz, m + 16, 32);
  v2f a0, a1;
  a0.x = hi ? sz0 : sx0;  a0.y = hi ? 0.0f : sy0;
  a1.x = hi ? sz1 : sx1;  a1.y = hi ? 0.0f : sy1;

  // |q|^2 of the 8 query rows this lane's C-VGPRs correspond to.
  float q2A[8], q2B[8];
#pragma unroll
  for (int r = 0; r < 8; ++r) {
    const int base = hi ? 8 : 0;
    q2A[r] = __shfl(q2, base + r, 32);
    q2B[r] = __shfl(q2, 16 + base + r, 32);
  }

  // ---- init top-k (keys = +inf) ----
#pragma unroll
  for (int e = 0; e < KNN; ++e) topk[w][e][lane] = ~0ull;
  unsigned long long curMax = ~0ull;
  int maxSlot = 0;

  // ---- stream candidate tiles: 2 f32 WMMAs per 16-candidate tile ----
  for (int tile = 0; tile < NPTS / 16; ++tile) {
    const int cidx = tile * 16 + m;
    const float cx = bp[(size_t)cidx * 3 + 0];
    const float cy = bp[(size_t)cidx * 3 + 1];
    const float cz = bp[(size_t)cidx * 3 + 2];
    const float c2 = cx * cx + cy * cy + cz * cz;

    v2f bb;                        // B is 4x16: V0 = {K0 | K2}, V1 = {K1 | K3}
    bb.x = hi ? cz : cx;
    bb.y = hi ? 0.0f : cy;

    v8f zero = {};
    v8f acc0 = __builtin_amdgcn_wmma_f32_16x16x4_f32(false, a0, false, bb,
                                                     (short)0, zero, false, false);
    v8f acc1 = __builtin_amdgcn_wmma_f32_16x16x4_f32(false, a1, false, bb,
                                                     (short)0, zero, false, false);

    // dist = |q|^2 + |c|^2 - 2 q.c  ; transpose through LDS
#pragma unroll
    for (int r = 0; r < 8; ++r) {
      const int M0 = (hi ? 8 : 0) + r;
      dtile[w][M0][m]      = fmaxf(q2A[r] + c2 - 2.0f * acc0[r], 0.0f);
      dtile[w][16 + M0][m] = fmaxf(q2B[r] + c2 - 2.0f * acc1[r], 0.0f);
    }
    asm volatile("" ::: "memory");   // per-wave DS is in-order; block compiler reordering

    // each lane consumes its own query row
#pragma unroll
    for (int n = 0; n < 16; ++n) {
      const float dd = dtile[w][lane][n];
      const unsigned long long key =
          ((unsigned long long)__float_as_uint(dd) << 32) |
          (unsigned int)(tile * 16 + n);
      if (key < curMax) {
        topk[w][maxSlot][lane] = key;
        curMax = 0ull;
        maxSlot = 0;
        for (int e = 0; e < KNN; ++e) {
          const unsigned long long v = topk[w][e][lane];
          if (v > curMax) { curMax = v; maxSlot = e; }
        }
      }
    }
    asm volatile("" ::: "memory");
  }

  // ---- sort this lane's 32 keys ascending (dist, then index) ----
  for (int i = 0; i < KNN - 1; ++i) {
    unsigned long long best = topk[w][i][lane];
    int bj = i;
    for (int j = i + 1; j < KNN; ++j) {
      const unsigned long long v = topk[w][j][lane];
      if (v < best) { best = v; bj = j; }
    }
    if (bj != i) {
      topk[w][bj][lane] = topk[w][i][lane];
      topk[w][i][lane]  = best;
    }
  }

  // ---- pass 1: neighborhood mean ----
  float sx = 0.0f, sy = 0.0f, sz = 0.0f;
#pragma unroll
  for (int j = 0; j < KNN; ++j) {
    const unsigned int id = (unsigned int)topk[w][j][lane];
    sx += bp[(size_t)id * 3 + 0];
    sy += bp[(size_t)id * 3 + 1];
    sz += bp[(size_t)id * 3 + 2];
  }
  const float inv = 1.0f / (float)KNN;
  const float ccx = sx * inv, ccy = sy * inv, ccz = sz * inv;

  const size_t pt = (size_t)b * NPTS + q;
  C_out[pt * 3 + 0] = ccx;
  C_out[pt * 3 + 1] = ccy;
  C_out[pt * 3 + 2] = ccz;

  // ---- pass 2: centered neighbors (output) + 3x3 covariance ----
  float rxx = 0, rxy = 0, rxz = 0, ryy = 0, ryz = 0, rzz = 0;
  float* __restrict__ pc = PC_out + pt * KNN * 3;
#pragma unroll
  for (int j = 0; j < KNN; ++j) {
    const unsigned int id = (unsigned int)topk[w][j][lane];
    const float dx = bp[(size_t)id * 3 + 0] - ccx;
    const float dy = bp[(size_t)id * 3 + 1] - ccy;
    const float dz = bp[(size_t)id * 3 + 2] - ccz;
    pc[j * 3 + 0] = dx;
    pc[j * 3 + 1] = dy;
    pc[j * 3 + 2] = dz;
    rxx += dx * dx; rxy += dx * dy; rxz += dx * dz;
    ryy += dy * dy; ryz += dy * dz; rzz += dz * dz;
  }

  // ---- 3x3 symmetric eigendecomposition (cyclic Jacobi, 6 sweeps) ----
  float Am[3][3] = {{rxx, rxy, rxz}, {rxy, ryy, ryz}, {rxz, ryz, rzz}};
  float Vm[3][3] = {{1.f, 0.f, 0.f}, {0.f, 1.f, 0.f}, {0.f, 0.f, 1.f}};
#pragma unroll
  for (int sweep = 0; sweep < 6; ++sweep) {
    JROTA(0, 1);
    JROTA(0, 2);
    JROTA(1, 2);
  }

  // sort eigenpairs ascending (match jnp.linalg.eigh)
  float e0 = Am[0][0], e1 = Am[1][1], e2 = Am[2][2];
  int i0 = 0, i1 = 1, i2 = 2;
  if (e0 > e1) { float t = e0; e0 = e1; e1 = t; int ti = i0; i0 = i1; i1 = ti; }
  if (e1 > e2) { float t = e1; e1 = e2; e2 = t; int ti = i1; i1 = i2; i2 = ti; }
  if (e0 > e1) { float t = e0; e0 = e1; e1 = t; int ti = i0; i0 = i1; i1 = ti; }

  float* __restrict__ Fo = F_out + pt * 9;
#pragma unroll
  for (int row = 0; row < 3; ++row) {
    Fo[row * 3 + 0] = pick3(Vm[row][0], Vm[row][1], Vm[row][2], i0);
    Fo[row * 3 + 1] = pick3(Vm[row][0], Vm[row][1], Vm[row][2], i1);
    Fo[row * 3 + 2] = pick3(Vm[row][0], Vm[row][1], Vm[row][2], i2);
  }
}

extern "C" void kernel_launch(void* const* d_in, const int* in_sizes, int n_in,
                              void* d_out, int out_size, void* d_ws, size_t ws_size,
                              hipStream_t stream) {
  (void)n_in; (void)d_ws; (void)ws_size; (void)out_size; (void)in_sizes;
  const float* pnts = (const float*)d_in[0];   // [8, 4096, 3] f32
  // d_in[1] is k == 32 (hardcoded as KNN)

  float* out   = (float*)d_out;
  float* F_out = out;                                              // [8,4096,3,3]
  float* C_out = out + (size_t)BATCH * NPTS * 9;                   // [8,4096,3]
  float* PC_out = out + (size_t)BATCH * NPTS * 9
                      + (size_t)BATCH * NPTS * 3;                  // [8,4096,32,3]

  const int total  = BATCH * NPTS;                 // 32768 points
  const int blocks = total / BLOCK_THREADS;        // 256
  knn_frames_kernel<<<blocks, BLOCK_THREADS, 0, stream>>>(pnts, F_out, C_out, PC_out);
}